// MultiLIF_43843026157719
// MI455X (gfx1250) — compile-verified
//
#include <hip/hip_runtime.h>
#include <stdint.h>

// Adaptive-LIF scan, MI455X (gfx1250).
// Memory-bound (402 MB @ 23.3 TB/s => ~17us floor; 0.026 FLOP/B -> WMMA N/A).
// CDNA5 path: global_load_async_to_lds_b128 (ASYNCcnt) with a 4-buffer LDS
// ring and depth-3 prefetch (~12 MB in flight chip-wide to cover BW*latency),
// s_wait_asynccnt pipelining, one s_barrier per chunk, NT streaming stores.

namespace {

constexpr int BB = 32;    // batch
constexpr int LL = 2048;  // timesteps
constexpr int KK = 512;   // neurons per batch row
constexpr int KB = 256;   // neurons handled per block (== blockDim.x)
constexpr int TC = 64;    // timesteps staged per LDS chunk (64 KB/chunk)
constexpr int NCH = LL / TC;          // 32 chunks
constexpr int NBUF = 4;               // LDS ring depth (256 KB of 320 KB WGP LDS)
constexpr int OPS = TC / 4;           // 16 async b128 ops per wave per chunk

// Issue one chunk's worth of async b128 loads: thread i = (r_off=i>>6, c=i&63)
// moves 16B of row (t0 + step*4 + r_off) into the ring buffer.
__device__ __forceinline__ void issue_chunk(const char* gbase, uint32_t lbase,
                                            uint32_t chunk, uint32_t r_off,
                                            uint32_t c16) {
    uint32_t voff  = chunk * (uint32_t)(TC * KK * 4) + r_off * 2048u + c16;
    uint32_t laddr = lbase + r_off * 1024u + c16;
    #pragma unroll
    for (int s = 0; s < OPS; ++s) {
        asm volatile("global_load_async_to_lds_b128 %0, %1, %2 th:TH_LOAD_NT"
                     :: "v"(laddr), "v"(voff), "s"(gbase) : "memory");
        voff  += 4u * 2048u;   // 4 rows * K*4 bytes (global)
        laddr += 4u * 1024u;   // 4 rows * KB*4 bytes (LDS)
    }
}

__global__ __launch_bounds__(KB)
void MultiLIF_43843026157719_kernel(const float* __restrict__ I,
                                    float* __restrict__ out) {
    __shared__ float lbuf[NBUF][TC * KB];   // 4 x 64 KB ring

    const int tid   = threadIdx.x;
    const int b     = blockIdx.x >> 1;
    const int kbase = (blockIdx.x & 1) * KB;

    const char* gbase =
        (const char*)(I + ((size_t)b * LL * KK + (size_t)kbase));

    // LDS byte offset = low 32 bits of flat pointer.
    const uint32_t lds0  = (uint32_t)(uintptr_t)(&lbuf[0][0]);
    const uint32_t r_off = (uint32_t)(tid >> 6);        // 0..3
    const uint32_t c16   = (uint32_t)(tid & 63) * 16u;  // byte col in row

    // ---- prologue: fill pipeline with chunks 0..2 (48 ops/wave <= 63) ----
    #pragma unroll
    for (uint32_t q = 0; q < NBUF - 1; ++q)
        issue_chunk(gbase, lds0 + q * (uint32_t)(TC * KB * 4), q, r_off, c16);

    // Neuron state for k = kbase + tid
    float v = 0.0f, a = 0.0f, n = 0.0f;

    float* op = out + ((size_t)b * LL * KK + (size_t)(kbase + tid));
    const size_t SOFF = (size_t)BB * LL * KK;  // spike_series offset

    for (int c = 0; c < NCH; ++c) {
        // Retire chunk c: async loads complete in order per wave, so allowing
        // only the ops issued after chunk c to remain outstanding suffices.
        if (c + 2 < NCH) {
            asm volatile("s_wait_asynccnt 0x20" ::: "memory"); // chunks c+1,c+2
        } else if (c + 1 < NCH) {
            asm volatile("s_wait_asynccnt 0x10" ::: "memory"); // chunk c+1
        } else {
            asm volatile("s_wait_asynccnt 0x0" ::: "memory");
        }
        // Rendezvous: (a) every wave's slice of chunk c is in LDS,
        // (b) every wave finished reading buf[(c+3)&3] (held chunk c-1).
        __syncthreads();

        // Deep prefetch: chunk c+3 into the buffer freed by chunk c-1.
        if (c + NBUF - 1 < NCH)
            issue_chunk(gbase,
                        lds0 + (uint32_t)((c + NBUF - 1) & (NBUF - 1)) *
                               (uint32_t)(TC * KB * 4),
                        (uint32_t)(c + NBUF - 1), r_off, c16);

        const float* lb = &lbuf[c & (NBUF - 1)][0];
        #pragma unroll 4
        for (int r = 0; r < TC; ++r) {
            float It = lb[r * KB + tid];                    // ds_load
            float th = __builtin_fmaf(1.5f, a, 1.0f);       // TH + BETA*a
            v = v - v * (1.0f / 20.0f) + It;                // leak + input
            bool  fire = (v >= th);
            float s = fire ? 1.0f : 0.0f;                   // s_hard == s fwd
            n += s;
            v = fire ? -0.5f : v;                           // reset (select)
            a = a - a * (1.0f / 100.0f) + s;                // adaptation
            __builtin_nontemporal_store(s, op);             // spikes
            __builtin_nontemporal_store(n, op + SOFF);      // spike_series
            op += KK;
        }
    }
}

} // namespace

extern "C" void kernel_launch(void* const* d_in, const int* in_sizes, int n_in,
                              void* d_out, int out_size, void* d_ws, size_t ws_size,
                              hipStream_t stream) {
    (void)in_sizes; (void)n_in; (void)out_size; (void)d_ws; (void)ws_size;
    const float* I = (const float*)d_in[0];
    float* out = (float*)d_out;
    dim3 grid(BB * (KK / KB));  // 64 blocks (one 256-neuron slice each)
    dim3 block(KB);             // 256 threads = 8 wave32
    MultiLIF_43843026157719_kernel<<<grid, block, 0, stream>>>(I, out);
}